// LinearRNNCell_21758304322014
// MI455X (gfx1250) — compile-verified
//
#include <hip/hip_runtime.h>

// Problem: h_t = h_{t-1} @ W_hh^T + x_t @ W_xh^T + b ; states[t] = h_t
// T=4096, B=32, D_IN=256, H=512, all fp32.
#define T_STEPS 4096
#define BATCH   32
#define DIN     256
#define HDIM    512

#define NWG     8                 // workgroups cooperating on the scan
#define NSLICE  (HDIM / NWG)      // 64 output columns per workgroup

// Padded LDS row strides (floats) to avoid bank conflicts on fragment reads
#define A1_STRIDE (DIN + 4)       // 260  (phase-1 A strip)
#define W_STRIDE  (HDIM + 4)      // 516  (W_hh^T slice, stored [nloc][k])
#define H_STRIDE  (HDIM + 4)      // 516  (h state, stored [b][k])

typedef __attribute__((ext_vector_type(2))) float v2f;
typedef __attribute__((ext_vector_type(8))) float v8f;

// ---------------------------------------------------------------------------
// CDNA5 async global->LDS DMA (VGLOBAL encoding, tracked by ASYNCcnt).
// vDST holds the LDS byte offset; the 64-bit global address rides a VGPR pair.
// Bypasses VGPRs entirely: cache/memory -> LDS.
// ---------------------------------------------------------------------------
__device__ __forceinline__ void async_g2l_b128(void* lds_dst, const void* gsrc)
{
  const unsigned loff =
      (unsigned)(unsigned long long)(__attribute__((address_space(3))) void*)lds_dst;
  asm volatile("global_load_async_to_lds_b128 %0, %1, off"
               :
               : "v"(loff), "v"(gsrc)
               : "memory");
}

__device__ __forceinline__ void wait_asynccnt0()
{
  asm volatile("s_wait_asynccnt 0x0" ::: "memory");
}

// ---------------------------------------------------------------------------
// WMMA f32 16x16x4 fragment layouts (ISA 7.12.2, wave32):
//   A (16x4):  lane l: M = l&15 ; VGPR0 = A[M][2*(l>>4)],   VGPR1 = A[M][2*(l>>4)+1]
//   B (4x16):  lane l: N = l&15 ; VGPR0 = B[2*(l>>4)][N],   VGPR1 = B[2*(l>>4)+1][N]
//   C/D:       VGPR i: lanes0-15 -> (M=i, N=lane), lanes16-31 -> (M=i+8, N=lane-16)
// Both A and B fragments are 2 consecutive K elements -> 8-byte loads.
// ---------------------------------------------------------------------------

// ============================================================================
// Phase 1: xproj[t*B+b][h] = inputs[t*B+b][:] . W_xh[h][:] + bias[h]
// GEMM M=131072, N=512, K=256, memory-bound (134 MB in + 268 MB out ~ 17 us
// at 23.3 TB/s). Written directly into d_out (same shape as states); phase 2
// consumes and overwrites it in place.
// ============================================================================
__global__ __launch_bounds__(256) void
xproj_wmma_kernel(const float* __restrict__ inp,   // [T*B, DIN]
                  const float* __restrict__ wxh,   // [HDIM, DIN]
                  const float* __restrict__ bias,  // [HDIM]
                  float* __restrict__ out)         // [T*B, HDIM]
{
  __shared__ float alds[16 * A1_STRIDE];           // 16 rows of A, padded

  const int mbase = blockIdx.x * 16;
  const int nblk  = blockIdx.y * 128;
  const int tid   = threadIdx.x;
  const int wave  = tid >> 5;
  const int lane  = tid & 31;

  // Async-stage the 16x256 A strip into LDS (16 B per lane per op).
  for (int idx = tid; idx < 16 * (DIN / 4); idx += 256) {
    const int r = idx >> 6;            // row 0..15
    const int c = idx & 63;            // float4 col 0..63
    async_g2l_b128(alds + r * A1_STRIDE + c * 4,
                   inp + (size_t)(mbase + r) * DIN + c * 4);
  }
  wait_asynccnt0();
  __syncthreads();

  const int nbase = nblk + wave * 16;
  const int row   = lane & 15;          // M (for A) / N (for B,C) inside tile
  const int kh    = lane >> 4;          // K half selector

  // C init: bias[col] — col depends only on lane, same for all 8 acc regs.
  const float bv = bias[nbase + row];
  v8f acc;
#pragma unroll
  for (int i = 0; i < 8; ++i) acc[i] = bv;

  const float* arow = alds + row * A1_STRIDE;              // A row (LDS)
  const float* brow = wxh + (size_t)(nbase + row) * DIN;   // B[k][n] = W_xh[n][k]

#pragma unroll 4
  for (int k = 0; k < DIN; k += 4) {
    v2f a = *(const v2f*)(arow + k + 2 * kh);
    v2f b = *(const v2f*)(brow + k + 2 * kh);
    acc = __builtin_amdgcn_wmma_f32_16x16x4_f32(
        /*neg_a=*/false, a, /*neg_b=*/false, b,
        /*c_mod=*/(short)0, acc, /*reuse_a=*/false, /*reuse_b=*/false);
  }

  const int col = nbase + row;
#pragma unroll
  for (int i = 0; i < 8; ++i) {
    const int m = mbase + i + 8 * kh;
    out[(size_t)m * HDIM + col] = acc[i];
  }
}

// ============================================================================
// Phase 2: sequential scan. NWG=8 persistent workgroups, 256 threads each.
// WG w owns output columns [w*64, w*64+64):
//   - W_hh^T slice (512x64 fp32) resident in LDS for all 4096 steps (132 KB)
//   - full h (32x512 fp32) async-DMA'd into LDS each step (66 KB)
//   - h exchanged through double-buffered global scratch (L2-resident, 64 KB)
//   - per-step device barrier: release fence -> atomicAdd(flags[t]) ->
//     acquire spin until == NWG
// Per wave per step: 128 x v_wmma_f32_16x16x4_f32 (K=512); acc is seeded from
// xproj (read from d_out[t]) and the result overwrites d_out[t] = states[t].
// ============================================================================
__global__ __launch_bounds__(256) void
rnn_scan_wmma_kernel(const float* __restrict__ whh,  // [HDIM, HDIM]
                     float* __restrict__ out,        // [T*B, HDIM] in: xproj, out: states
                     float* __restrict__ hbuf,       // ws: 2 * BATCH*HDIM floats
                     int* __restrict__ flags)        // ws: T_STEPS ints, zeroed
{
  __shared__ float wlds[NSLICE * W_STRIDE];          // [nloc][k], padded (132 KB)
  __shared__ float hlds[BATCH * H_STRIDE];           // [b][k],    padded (66 KB)

  const int wg    = blockIdx.x;
  const int nwgb  = wg * NSLICE;
  const int tid   = threadIdx.x;
  const int wave  = tid >> 5;
  const int lane  = tid & 31;

  // One-time async stage of the W_hh^T slice: wlds[nloc][k] = W_hh[nwgb+nloc][k]
  // (straight rows, reused for all 4096 steps).
  for (int idx = tid; idx < NSLICE * (HDIM / 4); idx += 256) {
    const int r = idx >> 7;            // nloc 0..63
    const int c = idx & 127;           // float4 col 0..127
    async_g2l_b128(wlds + r * W_STRIDE + c * 4,
                   whh + (size_t)(nwgb + r) * HDIM + c * 4);
  }
  wait_asynccnt0();
  __syncthreads();

  const int tile_m = (wave & 1) * 16;          // batch-row tile (0 or 16)
  const int tile_n = (wave >> 1) * 16;         // column tile inside slice
  const int row    = lane & 15;
  const int kh     = lane >> 4;
  const int colg   = nwgb + tile_n + row;      // global output column
  const int colw   = tile_n + row;             // column inside LDS W slice

  const float* arow = hlds + (tile_m + row) * H_STRIDE;
  const float* brow = wlds + colw * W_STRIDE;  // B[k][n] = W_hh[n][k]

  for (int t = 0; t < T_STEPS; ++t) {
    const float* hsrc = hbuf + (size_t)(t & 1) * (BATCH * HDIM);
    float*       hdst = hbuf + (size_t)((t + 1) & 1) * (BATCH * HDIM);

    // Async-DMA full h_t into LDS (64 KB from L2-resident scratch, no VGPR
    // round-trip; 16 b128 ops per thread, single ASYNCcnt wait).
    for (int idx = tid; idx < BATCH * (HDIM / 4); idx += 256) {
      const int r = idx >> 7;
      const int c = idx & 127;
      async_g2l_b128(hlds + r * H_STRIDE + c * 4,
                     hsrc + (size_t)r * HDIM + c * 4);
    }
    wait_asynccnt0();
    __syncthreads();

    // Seed accumulator with the xproj tile currently living in d_out[t].
    float* orow = out + (size_t)t * (BATCH * HDIM);
    v8f acc;
#pragma unroll
    for (int i = 0; i < 8; ++i)
      acc[i] = orow[(size_t)(tile_m + i + 8 * kh) * HDIM + colg];

    // h_t (LDS) @ W_hh^T slice (LDS): K=512 in steps of 4.
#pragma unroll 4
    for (int k = 0; k < HDIM; k += 4) {
      v2f a = *(const v2f*)(arow + k + 2 * kh);
      v2f b = *(const v2f*)(brow + k + 2 * kh);
      acc = __builtin_amdgcn_wmma_f32_16x16x4_f32(
          false, a, false, b, (short)0, acc, false, false);
    }

    // states[t] slice (overwrite xproj in place) + next-step h buffer.
#pragma unroll
    for (int i = 0; i < 8; ++i) {
      const int m = tile_m + i + 8 * kh;
      const float v = acc[i];
      orow[(size_t)m * HDIM + colg] = v;
      hdst[m * HDIM + colg] = v;
    }

    // ---- device-scope split barrier over the 8 workgroups ----
    __syncthreads();                                   // all stores issued
    __builtin_amdgcn_fence(__ATOMIC_RELEASE, "agent"); // make them visible
    if (tid == 0) {
      __hip_atomic_fetch_add(&flags[t], 1, __ATOMIC_RELEASE,
                             __HIP_MEMORY_SCOPE_AGENT);
      while (__hip_atomic_load(&flags[t], __ATOMIC_ACQUIRE,
                               __HIP_MEMORY_SCOPE_AGENT) < NWG) {
      }
    }
    __syncthreads();
    __builtin_amdgcn_fence(__ATOMIC_ACQUIRE, "agent"); // invalidate L0 for h reload
  }
}

// Zero the per-step barrier flags and seed h buffer 0 with h0.
__global__ void scan_init_kernel(const float* __restrict__ h0,
                                 float* __restrict__ hbuf0,
                                 int* __restrict__ flags)
{
  const int i = blockIdx.x * blockDim.x + threadIdx.x;
  if (i < BATCH * HDIM) hbuf0[i] = h0[i];
  if (i < T_STEPS) flags[i] = 0;
}

extern "C" void kernel_launch(void* const* d_in, const int* in_sizes, int n_in,
                              void* d_out, int out_size, void* d_ws, size_t ws_size,
                              hipStream_t stream)
{
  const float* inputs = (const float*)d_in[0];  // [T,B,DIN]
  const float* h0     = (const float*)d_in[1];  // [B,H]
  const float* whh    = (const float*)d_in[2];  // [H,H]
  const float* wxh    = (const float*)d_in[3];  // [H,DIN]
  const float* bias   = (const float*)d_in[4];  // [H]
  float* out = (float*)d_out;                   // [T,B,H]

  // Workspace layout: h double buffer (2*64 KB) + per-step flags (16 KB).
  float* hbuf  = (float*)d_ws;
  int*   flags = (int*)(hbuf + 2 * BATCH * HDIM);

  scan_init_kernel<<<(BATCH * HDIM + 255) / 256, 256, 0, stream>>>(h0, hbuf, flags);

  // Phase 1: xproj -> d_out.  Grid: 8192 M-tiles x 4 N-strips of 128.
  dim3 g1((T_STEPS * BATCH) / 16, HDIM / 128);
  xproj_wmma_kernel<<<g1, 256, 0, stream>>>(inputs, wxh, bias, out);

  // Phase 2: persistent cooperative scan, 8 workgroups.
  rnn_scan_wmma_kernel<<<NWG, 256, 0, stream>>>(whh, out, hbuf, flags);
}